// ConformerSelfAttention_48825188220944
// MI455X (gfx1250) — compile-verified
//
#include <hip/hip_runtime.h>
#include <hip/hip_bf16.h>

// CDNA5 (gfx1250) fp32 WMMA types
typedef __attribute__((ext_vector_type(2))) float v2f;
typedef __attribute__((ext_vector_type(8))) float v8f;
// 16-byte payload type expected by the async-LDS builtin (per hipcc diagnostic)
typedef int v4i __attribute__((vector_size(16)));

#define HIDDEN   1024
#define NHEADS   16
#define HEAD     64
#define SEQ      2048
#define BATCH    4
#define LEFT     64
#define RIGHT    8
#define NUM_POS  73      // LEFT + RIGHT + 1
#define MROWS    8192    // BATCH * SEQ

// ---------------------------------------------------------------------------
// Async global->LDS staging (GLOBAL_LOAD_ASYNC_TO_LDS_B128, ASYNCcnt-tracked).
// Falls back to float4 register staging if the builtin is absent.
// ---------------------------------------------------------------------------
#if defined(__gfx1250__) && __has_builtin(__builtin_amdgcn_global_load_async_to_lds_b128)
#define HAS_ASYNC_LDS 1
#else
#define HAS_ASYNC_LDS 0
#endif

__device__ __forceinline__ void stage16(const float* __restrict__ g,
                                        float* __restrict__ l) {
#if HAS_ASYNC_LDS
    // flat LDS addresses carry the LDS offset in addr[31:0] (aperture rule),
    // so int-casts to AS(1)/AS(3) v4i pointers produce the operand kinds the
    // builtin wants (per compiler diagnostic: pointer to 16B int vector).
    __builtin_amdgcn_global_load_async_to_lds_b128(
        (__attribute__((address_space(1))) v4i*)(size_t)g,
        (__attribute__((address_space(3))) v4i*)(size_t)l,
        0, 0);
#else
    *(float4*)l = *(const float4*)g;
#endif
}

__device__ __forceinline__ void stage_drain() {
#if HAS_ASYNC_LDS
#if __has_builtin(__builtin_amdgcn_s_wait_asynccnt)
    __builtin_amdgcn_s_wait_asynccnt(0);
#else
    asm volatile("s_wait_asynccnt 0x0" ::: "memory");
#endif
#endif
}

// ---------------------------------------------------------------------------
// GEMM: Y[M,N] = X[M,K] @ W[N,K]^T + bias[N]   (torch Linear convention)
// Block tile 64(M) x 128(N), BK=16, 128 threads = 4 waves.
// Each wave: 64x32 sub-tile = 4(M) x 2(N) WMMA 16x16 tiles, fp32 v_wmma.
// ---------------------------------------------------------------------------
#define GPAD 20  // 16 + 4 pad: rows 16B-aligned, rows hit distinct LDS banks

__global__ __launch_bounds__(128) void gemm_xwt_bias(
    const float* __restrict__ X, const float* __restrict__ W,
    const float* __restrict__ bias, float* __restrict__ Y,
    int M, int N, int K)
{
    __shared__ float Xs[64 * GPAD];
    __shared__ float Ws[128 * GPAD];

    const int tid  = threadIdx.x;
    const int lane = tid & 31;
    const int wave = tid >> 5;
    const int hl   = lane >> 4;   // half-wave selector
    const int l15  = lane & 15;
    const int m0   = blockIdx.y * 64;
    const int n0   = blockIdx.x * 128;
    const int wn0  = wave * 32;   // wave's column offset inside block tile

    v8f zero = {};
    v8f acc[4][2];
    for (int mt = 0; mt < 4; ++mt)
        for (int nt = 0; nt < 2; ++nt) acc[mt][nt] = zero;

    for (int k0 = 0; k0 < K; k0 += 16) {
        __syncthreads();
        // stage X tile 64x16 (256 float4, 2 per thread)
        #pragma unroll
        for (int i = 0; i < 2; ++i) {
            int id = tid * 2 + i;
            int r = id >> 2, c4 = (id & 3) * 4;
            stage16(X + (size_t)(m0 + r) * K + k0 + c4, Xs + r * GPAD + c4);
        }
        // stage W tile 128x16 (512 float4, 4 per thread)
        #pragma unroll
        for (int i = 0; i < 4; ++i) {
            int id = tid * 4 + i;
            int r = id >> 2, c4 = (id & 3) * 4;
            stage16(W + (size_t)(n0 + r) * K + k0 + c4, Ws + r * GPAD + c4);
        }
        stage_drain();
        __syncthreads();

        #pragma unroll
        for (int ks = 0; ks < 16; ks += 4) {
            const int kk = ks + 2 * hl;       // fragment k: kk + v, v in {0,1}
            v2f a[4], b[2];
            #pragma unroll
            for (int mt = 0; mt < 4; ++mt) {
                a[mt][0] = Xs[(mt * 16 + l15) * GPAD + kk];
                a[mt][1] = Xs[(mt * 16 + l15) * GPAD + kk + 1];
            }
            #pragma unroll
            for (int nt = 0; nt < 2; ++nt) {
                b[nt][0] = Ws[(wn0 + nt * 16 + l15) * GPAD + kk];
                b[nt][1] = Ws[(wn0 + nt * 16 + l15) * GPAD + kk + 1];
            }
            #pragma unroll
            for (int mt = 0; mt < 4; ++mt)
                #pragma unroll
                for (int nt = 0; nt < 2; ++nt)
                    acc[mt][nt] = __builtin_amdgcn_wmma_f32_16x16x4_f32(
                        false, a[mt], false, b[nt], (short)0, acc[mt][nt],
                        false, false);
        }
    }

    // epilogue: C/D layout  m = r + 8*hl,  n = l15
    #pragma unroll
    for (int nt = 0; nt < 2; ++nt) {
        const int col = n0 + wn0 + nt * 16 + l15;
        const float bv = bias[col];
        #pragma unroll
        for (int mt = 0; mt < 4; ++mt)
            #pragma unroll
            for (int r = 0; r < 8; ++r) {
                const int row = m0 + mt * 16 + r + 8 * hl;
                Y[(size_t)row * N + col] = acc[mt][nt][r] + bv;
            }
    }
}

// ---------------------------------------------------------------------------
// Flash-style attention with relative-key bias.
// One block per (b, h, 64-query tile); 128 threads = 4 waves.
// Wave w owns query rows [w*16, w*16+16) of the tile.
// S = Q.K^T (fp32 WMMA), fused scale + qd-bias, online softmax in LDS,
// O += P.V (fp32 WMMA). S/P tile reuses the K-tile LDS buffer.
// ---------------------------------------------------------------------------
#define APAD 68   // 64 + 4: rows 16B-aligned, distinct banks
#define QPAD 76   // 73 -> 76

__global__ __launch_bounds__(128) void attn_kernel(
    const float* __restrict__ Qb, const float* __restrict__ Kb,
    const float* __restrict__ Vb, const float* __restrict__ dist_emb,
    float* __restrict__ Ob)
{
    __shared__ float Qs[64 * APAD];
    __shared__ float KS[64 * APAD];   // K tile; reused as S/P tile
    __shared__ float Vs[64 * APAD];
    __shared__ float qd[64 * QPAD];   // scale * (q . dist_emb[p])
    __shared__ float m_i[64], l_i[64], alph[64];

    const int tid  = threadIdx.x;
    const int lane = tid & 31;
    const int wave = tid >> 5;
    const int hl   = lane >> 4;
    const int l15  = lane & 15;
    const int w16  = wave * 16;

    const int mt = blockIdx.x;        // query tile (0..31)
    const int h  = blockIdx.y;        // head
    const int b  = blockIdx.z;        // batch
    const float scale = 0.125f;       // 1/sqrt(64)

    const float* Q  = Qb + ((size_t)b * SEQ + mt * 64) * HIDDEN + h * HEAD;
    const float* Kp = Kb + (size_t)b * SEQ * HIDDEN + h * HEAD;
    const float* Vp = Vb + (size_t)b * SEQ * HIDDEN + h * HEAD;
    float*       Op = Ob + ((size_t)b * SEQ + mt * 64) * HIDDEN + h * HEAD;

    // Load Q tile 64x64 (1024 float4, 8 per thread)
    #pragma unroll
    for (int i = 0; i < 8; ++i) {
        int id = tid * 8 + i;
        int r = id >> 4, c4 = (id & 15) * 4;
        stage16(Q + (size_t)r * HIDDEN + c4, Qs + r * APAD + c4);
    }
    if (tid < 64) { m_i[tid] = -3.0e38f; l_i[tid] = 0.0f; }
    stage_drain();
    __syncthreads();

    // qd[r][p] = scale * dot(Q[r,:], dist_emb[p,:])   (64*73 small dots)
    for (int t = tid; t < 64 * NUM_POS; t += 128) {
        int r = t / NUM_POS, p = t - r * NUM_POS;
        const float* de = dist_emb + p * HEAD;
        float s = 0.0f;
        #pragma unroll 8
        for (int d = 0; d < HEAD; ++d) s += Qs[r * APAD + d] * de[d];
        qd[r * QPAD + p] = s * scale;
    }

    v8f zero = {};
    v8f o[4]; for (int nt = 0; nt < 4; ++nt) o[nt] = zero;

    for (int j = 0; j < SEQ / 64; ++j) {
        __syncthreads();   // prior P.V done with KS/Vs
        // stage K and V tiles (each 1024 float4) via async LDS DMA
        #pragma unroll
        for (int i = 0; i < 8; ++i) {
            int id = tid * 8 + i;
            int r = id >> 4, c4 = (id & 15) * 4;
            stage16(Kp + (size_t)(j * 64 + r) * HIDDEN + c4, KS + r * APAD + c4);
            stage16(Vp + (size_t)(j * 64 + r) * HIDDEN + c4, Vs + r * APAD + c4);
        }
        if (j + 1 < SEQ / 64) {  // prefetch next tiles into caches
            __builtin_prefetch(Kp + (size_t)((j + 1) * 64 + (tid >> 1)) * HIDDEN, 0, 1);
            __builtin_prefetch(Vp + (size_t)((j + 1) * 64 + (tid >> 1)) * HIDDEN, 0, 1);
        }
        stage_drain();
        __syncthreads();

        // ---- S = Q . K^T : wave rows [w16, w16+16) x 64 cols ----
        v8f s[4]; for (int nt = 0; nt < 4; ++nt) s[nt] = zero;
        #pragma unroll
        for (int ks = 0; ks < HEAD; ks += 4) {
            const int kk = ks + 2 * hl;
            v2f a;
            a[0] = Qs[(w16 + l15) * APAD + kk];
            a[1] = Qs[(w16 + l15) * APAD + kk + 1];
            #pragma unroll
            for (int nt = 0; nt < 4; ++nt) {
                v2f bf;     // B = K^T: element (k,n) = Ktile[n][k]
                bf[0] = KS[(nt * 16 + l15) * APAD + kk];
                bf[1] = KS[(nt * 16 + l15) * APAD + kk + 1];
                s[nt] = __builtin_amdgcn_wmma_f32_16x16x4_f32(
                    false, a, false, bf, (short)0, s[nt], false, false);
            }
        }
        __syncthreads();   // all waves done reading KS

        // ---- fused scale + relative bias, store S into KS ----
        #pragma unroll
        for (int nt = 0; nt < 4; ++nt)
            #pragma unroll
            for (int r = 0; r < 8; ++r) {
                const int m_loc = w16 + r + 8 * hl;
                const int n_loc = nt * 16 + l15;
                int dist = (j * 64 + n_loc) - (mt * 64 + m_loc);
                dist = dist < -LEFT ? -LEFT : (dist > RIGHT ? RIGHT : dist);
                KS[m_loc * APAD + n_loc] =
                    s[nt][r] * scale + qd[m_loc * QPAD + dist + LEFT];
            }
        __syncthreads();

        // ---- online softmax (one thread per query row) ----
        if (tid < 64) {
            float* row = KS + tid * APAD;
            float rmax = -3.0e38f;
            for (int n = 0; n < 64; ++n) rmax = fmaxf(rmax, row[n]);
            const float nm = fmaxf(m_i[tid], rmax);
            float sum = 0.0f;
            for (int n = 0; n < 64; ++n) {
                float p = __expf(row[n] - nm);
                row[n] = p; sum += p;
            }
            const float al = __expf(m_i[tid] - nm);
            l_i[tid] = l_i[tid] * al + sum;
            m_i[tid] = nm;
            alph[tid] = al;
        }
        __syncthreads();

        // ---- rescale O, then O += P . V ----
        #pragma unroll
        for (int r = 0; r < 8; ++r) {
            const float al = alph[w16 + r + 8 * hl];
            #pragma unroll
            for (int nt = 0; nt < 4; ++nt) o[nt][r] *= al;
        }
        #pragma unroll
        for (int ks = 0; ks < 64; ks += 4) {
            const int kk = ks + 2 * hl;
            v2f a;
            a[0] = KS[(w16 + l15) * APAD + kk];
            a[1] = KS[(w16 + l15) * APAD + kk + 1];
            #pragma unroll
            for (int nt = 0; nt < 4; ++nt) {
                v2f bf;     // B = Vtile: element (k,n) = Vs[k][n]
                bf[0] = Vs[kk * APAD + nt * 16 + l15];
                bf[1] = Vs[(kk + 1) * APAD + nt * 16 + l15];
                o[nt] = __builtin_amdgcn_wmma_f32_16x16x4_f32(
                    false, a, false, bf, (short)0, o[nt], false, false);
            }
        }
    }

    __syncthreads();
    #pragma unroll
    for (int r = 0; r < 8; ++r) {
        const int m_loc = w16 + r + 8 * hl;
        const float inv = 1.0f / l_i[m_loc];
        #pragma unroll
        for (int nt = 0; nt < 4; ++nt)
            Op[(size_t)m_loc * HIDDEN + nt * 16 + l15] = o[nt][r] * inv;
    }
}

// ---------------------------------------------------------------------------
extern "C" void kernel_launch(void* const* d_in, const int* in_sizes, int n_in,
                              void* d_out, int out_size, void* d_ws, size_t ws_size,
                              hipStream_t stream) {
    const float* hidden = (const float*)d_in[0];
    const float* Wq = (const float*)d_in[1];
    const float* bq = (const float*)d_in[2];
    const float* Wk = (const float*)d_in[3];
    const float* bk = (const float*)d_in[4];
    const float* Wv = (const float*)d_in[5];
    const float* bv = (const float*)d_in[6];
    const float* Wo = (const float*)d_in[7];
    const float* bo = (const float*)d_in[8];
    const float* dist_emb = (const float*)d_in[9];
    float* out = (float*)d_out;

    const size_t bufElems = (size_t)MROWS * HIDDEN;   // 8192*1024 floats
    float* qbuf = (float*)d_ws;                       // also attention output
    float* kbuf = qbuf + bufElems;
    float* vbuf = kbuf + bufElems;                    // ws needed: 96 MB

    dim3 ggrid(HIDDEN / 128, MROWS / 64);             // (8, 128)
    gemm_xwt_bias<<<ggrid, 128, 0, stream>>>(hidden, Wq, bq, qbuf, MROWS, HIDDEN, HIDDEN);
    gemm_xwt_bias<<<ggrid, 128, 0, stream>>>(hidden, Wk, bk, kbuf, MROWS, HIDDEN, HIDDEN);
    gemm_xwt_bias<<<ggrid, 128, 0, stream>>>(hidden, Wv, bv, vbuf, MROWS, HIDDEN, HIDDEN);

    // each block reads only its own Q rows before writing them -> alias O=Q
    dim3 agrid(SEQ / 64, NHEADS, BATCH);              // (32, 16, 4)
    attn_kernel<<<agrid, 128, 0, stream>>>(qbuf, kbuf, vbuf, dist_emb, qbuf);

    gemm_xwt_bias<<<ggrid, 128, 0, stream>>>(qbuf, Wo, bo, out, MROWS, HIDDEN, HIDDEN);
}